// GraphSAGE_45294725104183
// MI455X (gfx1250) — compile-verified
//
#include <hip/hip_runtime.h>

typedef __attribute__((ext_vector_type(2))) float v2f;
typedef __attribute__((ext_vector_type(4))) float f4v;
typedef __attribute__((ext_vector_type(8))) float v8f;

#define HID 128
#define FAN 16
#define WAVES 8
#define THREADS (WAVES * 32)

// LDS: weights Wcat[n][k] n in [0,128), k in [0,256)  (k<128: Wself, k>=128: Wneigh)
//      + per-wave neighbor-mean tile [16][128]
#define WLDS_FLOATS (HID * 256)
#define XLDS_FLOATS (WAVES * 16 * HID)
#define LDS_BYTES ((WLDS_FLOATS + XLDS_FLOATS) * 4)

// out[d,:] = act( self(d) @ Wself^T + bself + mean_k nbr(d,k) @ Wneigh^T + bneigh )
// INDIR: row r of the feature matrix is feat[map[r]]; else feat[r].
template <bool INDIR, bool RELU>
__global__ __launch_bounds__(THREADS) void sage_layer_kernel(
    const float* __restrict__ feat, const int* __restrict__ map,
    const int* __restrict__ nbr, const float* __restrict__ Wself,
    const float* __restrict__ bself, const float* __restrict__ Wneigh,
    const float* __restrict__ bneigh, float* __restrict__ out, int num_dst) {
  extern __shared__ float smem[];
  float* wlds = smem;                  // [128][256]
  float* xlds = smem + WLDS_FLOATS;    // [WAVES][16][128]

  const int tid = threadIdx.x;
  const int wave = tid >> 5;
  const int lane = tid & 31;
  const int lm = lane & 15;   // row (A) or col (B,C/D) within 16
  const int lh = lane >> 4;   // K-half selector for A/B fragments

  // Cooperative weight load into LDS: wlds[n][k] / wlds[n][128+k]
  {
    const f4v* ws4 = (const f4v*)Wself;
    const f4v* wn4 = (const f4v*)Wneigh;
    f4v* wl4 = (f4v*)wlds;
    for (int i = tid; i < HID * (HID / 4); i += THREADS) {
      int n = i >> 5;        // 32 float4 per 128-float weight row
      int kq = i & 31;
      wl4[n * 64 + kq] = ws4[i];
      wl4[n * 64 + 32 + kq] = wn4[i];
    }
  }
  __syncthreads();

  float* xw = xlds + wave * 16 * HID;
  const int ntiles = num_dst >> 4;
  for (int tile = blockIdx.x * WAVES + wave; tile < ntiles;
       tile += gridDim.x * WAVES) {
    // ---- neighbor gather + mean into LDS (cross-lane transpose needed) ----
    // Unroll 2 rows at a time: ~32 independent nbr->map->row chains in flight
    // per wave, so exposed HBM latency stays below the bandwidth floor.
#pragma unroll 2
    for (int m = 0; m < 16; ++m) {
      int d = tile * 16 + m;
      f4v a;
      a.x = a.y = a.z = a.w = 0.0f;
#pragma unroll
      for (int k = 0; k < FAN; ++k) {
        int nb = nbr[d * FAN + k];
        long rr = INDIR ? (long)map[nb] : (long)nb;
        f4v v = ((const f4v*)(feat + rr * HID))[lane];
        a.x += v.x; a.y += v.y; a.z += v.z; a.w += v.w;
      }
      const float inv = 1.0f / (float)FAN;
      a.x *= inv; a.y *= inv; a.z *= inv; a.w *= inv;
      ((f4v*)(xw + m * HID))[lane] = a;
    }

    // ---- WMMA: D[16x128] = [self | mean] (16x256) x Wcat^T (256x128) ----
    long srow = INDIR ? (long)map[tile * 16 + lm] : (long)(tile * 16 + lm);
    const float* selfrow = feat + srow * HID;

    v8f acc[8] = {};
#pragma unroll 4
    for (int k0 = 0; k0 < 128; k0 += 4) {  // self half: A direct from global
      v2f a = *(const v2f*)(selfrow + k0 + lh * 2);
#pragma unroll
      for (int nt = 0; nt < 8; ++nt) {
        v2f b = *(const v2f*)(wlds + (nt * 16 + lm) * 256 + k0 + lh * 2);
        acc[nt] = __builtin_amdgcn_wmma_f32_16x16x4_f32(
            false, a, false, b, (short)0, acc[nt], false, false);
      }
    }
#pragma unroll 4
    for (int k0 = 128; k0 < 256; k0 += 4) {  // neighbor half: A from LDS
      v2f a = *(const v2f*)(xw + lm * HID + (k0 - 128) + lh * 2);
#pragma unroll
      for (int nt = 0; nt < 8; ++nt) {
        v2f b = *(const v2f*)(wlds + (nt * 16 + lm) * 256 + k0 + lh * 2);
        acc[nt] = __builtin_amdgcn_wmma_f32_16x16x4_f32(
            false, a, false, b, (short)0, acc[nt], false, false);
      }
    }

    // ---- epilogue: bias (+ReLU) and store ----
#pragma unroll
    for (int nt = 0; nt < 8; ++nt) {
      int n = nt * 16 + lm;
      float bb = bself[n] + bneigh[n];
#pragma unroll
      for (int j = 0; j < 8; ++j) {
        int row = tile * 16 + j + lh * 8;  // C/D layout: vgpr j -> M=j (+8 hi half)
        float v = acc[nt][j] + bb;
        if (RELU) v = fmaxf(v, 0.0f);
        out[(long)row * HID + n] = v;
      }
    }
  }
}

// out[4096,64] = X[4096,128] @ W[64,128]^T + b   (weights stay hot in cache)
__global__ __launch_bounds__(THREADS) void out_proj_kernel(
    const float* __restrict__ X, const float* __restrict__ W,
    const float* __restrict__ b, float* __restrict__ out, int rows) {
  const int tid = threadIdx.x;
  const int wave = tid >> 5;
  const int lane = tid & 31;
  const int lm = lane & 15;
  const int lh = lane >> 4;
  const int ntiles = rows >> 4;
  for (int tile = blockIdx.x * WAVES + wave; tile < ntiles;
       tile += gridDim.x * WAVES) {
    const float* xrow = X + (long)(tile * 16 + lm) * HID;
    v8f acc[4] = {};
#pragma unroll 4
    for (int k0 = 0; k0 < HID; k0 += 4) {
      v2f a = *(const v2f*)(xrow + k0 + lh * 2);
#pragma unroll
      for (int nt = 0; nt < 4; ++nt) {
        v2f bf = *(const v2f*)(W + (nt * 16 + lm) * HID + k0 + lh * 2);
        acc[nt] = __builtin_amdgcn_wmma_f32_16x16x4_f32(
            false, a, false, bf, (short)0, acc[nt], false, false);
      }
    }
#pragma unroll
    for (int nt = 0; nt < 4; ++nt) {
      int n = nt * 16 + lm;
      float bb = b[n];
#pragma unroll
      for (int j = 0; j < 8; ++j) {
        int row = tile * 16 + j + lh * 8;
        out[(long)row * 64 + n] = acc[nt][j] + bb;
      }
    }
  }
}

extern "C" void kernel_launch(void* const* d_in, const int* in_sizes, int n_in,
                              void* d_out, int out_size, void* d_ws,
                              size_t ws_size, hipStream_t stream) {
  const float* emb = (const float*)d_in[0];
  const float* Wself0 = (const float*)d_in[1];
  const float* bself0 = (const float*)d_in[2];
  const float* Wneigh0 = (const float*)d_in[3];
  const float* bneigh0 = (const float*)d_in[4];
  const float* Wself1 = (const float*)d_in[5];
  const float* bself1 = (const float*)d_in[6];
  const float* Wneigh1 = (const float*)d_in[7];
  const float* bneigh1 = (const float*)d_in[8];
  const float* Wout = (const float*)d_in[9];
  const float* bout = (const float*)d_in[10];
  const int* input_nodes = (const int*)d_in[11];
  const int* nbr0 = (const int*)d_in[12];
  const int* nbr1 = (const int*)d_in[13];

  const int N_DST0 = 65536, N_DST1 = 4096;
  float* h0 = (float*)d_ws;                  // [65536,128]
  float* h1 = h0 + (size_t)N_DST0 * HID;     // [4096,128]

  // Opt in to >64KB dynamic LDS (no-op if already allowed; not a stream op).
  (void)hipFuncSetAttribute(
      reinterpret_cast<const void*>(&sage_layer_kernel<true, true>),
      hipFuncAttributeMaxDynamicSharedMemorySize, (int)LDS_BYTES);
  (void)hipFuncSetAttribute(
      reinterpret_cast<const void*>(&sage_layer_kernel<false, false>),
      hipFuncAttributeMaxDynamicSharedMemorySize, (int)LDS_BYTES);

  // Layer 0: 4096 row-tiles, 8 per block
  sage_layer_kernel<true, true><<<(N_DST0 / 16) / WAVES, THREADS, LDS_BYTES,
                                  stream>>>(emb, input_nodes, nbr0, Wself0,
                                            bself0, Wneigh0, bneigh0, h0,
                                            N_DST0);
  // Layer 1: 256 row-tiles
  sage_layer_kernel<false, false><<<(N_DST1 / 16) / WAVES, THREADS, LDS_BYTES,
                                    stream>>>(h0, nullptr, nbr1, Wself1, bself1,
                                              Wneigh1, bneigh1, h1, N_DST1);
  // Output projection
  out_proj_kernel<<<(N_DST1 / 16) / WAVES, THREADS, 0, stream>>>(h1, Wout, bout,
                                                                 (float*)d_out,
                                                                 N_DST1);
}